// TFSparseEmbedding_76828374991706
// MI455X (gfx1250) — compile-verified
//
#include <hip/hip_runtime.h>

typedef __attribute__((ext_vector_type(2))) float v2f;
typedef __attribute__((ext_vector_type(8))) float v8f;

#define SEGS_PER_GROUP 16
#define WAVES_PER_GROUP 4
#define DIM 64
#define CHUNKS 4            // DIM / 16

// One block = one group of 16 consecutive segments, 4 waves cooperating.
// Each wave accumulates a partial 16x64 tile via one-hot WMMA, then the
// partials are reduced through LDS (deterministic order, no atomics).
__global__ __launch_bounds__(32 * WAVES_PER_GROUP)
void seg_embed_sum_wmma(const float* __restrict__ params,
                        const int* __restrict__ ids,
                        const int* __restrict__ seg_ids,
                        float* __restrict__ out,
                        int n_ids, int num_segments)
{
    const int lane = threadIdx.x & 31;
    const int wave = threadIdx.x >> 5;
    const int group = blockIdx.x;
    const int seg_lo = group * SEGS_PER_GROUP;
    const int seg_hi = seg_lo + SEGS_PER_GROUP;

    // lower_bound over sorted segment ids (uniform across the wave)
    int beg, end;
    {
        int lo = 0, hi = n_ids;
        while (lo < hi) { int mid = (lo + hi) >> 1; if (seg_ids[mid] < seg_lo) lo = mid + 1; else hi = mid; }
        beg = lo;
        lo = beg; hi = n_ids;
        while (lo < hi) { int mid = (lo + hi) >> 1; if (seg_ids[mid] < seg_hi) lo = mid + 1; else hi = mid; }
        end = lo;
    }

    const int m  = lane & 15;   // A: row M.  B/C/D: column N.
    const int kh = lane >> 4;   // K half: lanes 0-15 -> K{0,1}, lanes 16-31 -> K{2,3}

    v8f acc[CHUNKS];
#pragma unroll
    for (int c = 0; c < CHUNKS; ++c) acc[c] = (v8f)0.0f;

    // Waves stride over 4-id chunks of this group's id range.
    for (int base = beg + wave * 4; base < end; base += WAVES_PER_GROUP * 4) {
        // This lane supplies K slots k0 = 2*kh and k1 = 2*kh + 1.
        const int i0 = base + 2 * kh;
        const int i1 = i0 + 1;
        const int ci0 = (i0 < end) ? i0 : (end - 1);   // clamp: safe loads for tail
        const int ci1 = (i1 < end) ? i1 : (end - 1);

        const int id0 = ids[ci0];
        const int id1 = ids[ci1];
        const int s0  = seg_ids[ci0] - seg_lo;
        const int s1  = seg_ids[ci1] - seg_lo;

        // One-hot A matrix: A[m][k] = 1 iff id_k belongs to local segment m.
        v2f A;
        A.x = ((i0 < end) && (s0 == m)) ? 1.0f : 0.0f;
        A.y = ((i1 < end) && (s1 == m)) ? 1.0f : 0.0f;

        const float* __restrict__ r0 = params + (size_t)id0 * DIM;
        const float* __restrict__ r1 = params + (size_t)id1 * DIM;

#pragma unroll
        for (int c = 0; c < CHUNKS; ++c) {
            // B[k][n]: lane n<16 half holds K{0,1}, upper half K{2,3}.
            v2f B;
            B.x = r0[c * 16 + m];
            B.y = r1[c * 16 + m];
            // D = A*B + C  (8 args: neg_a, A, neg_b, B, c_mod, C, reuse_a, reuse_b)
            acc[c] = __builtin_amdgcn_wmma_f32_16x16x4_f32(
                false, A, false, B, (short)0, acc[c], false, false);
        }
    }

    // Cross-wave reduction of the 4 partial C tiles through LDS.
    // Pad inner dim to 33 floats so lane stride (33) is coprime with 64 banks.
    __shared__ float red[WAVES_PER_GROUP][32][CHUNKS * 8 + 1];
#pragma unroll
    for (int c = 0; c < CHUNKS; ++c)
#pragma unroll
        for (int v = 0; v < 8; ++v)
            red[wave][lane][c * 8 + v] = acc[c][v];
    __syncthreads();

    // Wave w finalizes dim-chunk c = w (WAVES_PER_GROUP == CHUNKS).
    {
        const int c = wave;
#pragma unroll
        for (int v = 0; v < 8; ++v) {
            float s = red[0][lane][c * 8 + v];
#pragma unroll
            for (int w2 = 1; w2 < WAVES_PER_GROUP; ++w2)
                s += red[w2][lane][c * 8 + v];
            const int row = seg_lo + v + 8 * kh;       // C/D layout: VGPR v -> M=v / M=v+8
            if (row < num_segments)
                out[(size_t)row * DIM + c * 16 + m] = s;  // 64B-coalesced per half-wave
        }
    }
}

extern "C" void kernel_launch(void* const* d_in, const int* in_sizes, int n_in,
                              void* d_out, int out_size, void* d_ws, size_t ws_size,
                              hipStream_t stream) {
    const float* params   = (const float*)d_in[0];
    const int*   ids      = (const int*)d_in[1];
    const int*   seg_ids  = (const int*)d_in[2];
    float*       out      = (float*)d_out;

    const int n_ids        = in_sizes[1];
    const int num_segments = out_size / DIM;
    const int groups       = (num_segments + SEGS_PER_GROUP - 1) / SEGS_PER_GROUP;

    seg_embed_sum_wmma<<<dim3(groups), dim3(32 * WAVES_PER_GROUP), 0, stream>>>(
        params, ids, seg_ids, out, n_ids, num_segments);
}